// CNN_PHMM_VAE_45337674777012
// MI455X (gfx1250) — compile-verified
//
#include <hip/hip_runtime.h>
#include <hip/hip_bf16.h>

#define NEGV (-100.0f)
#define LOG_QUARTER (-1.3862943611198906f)

#define AS1 __attribute__((address_space(1)))
#define AS3 __attribute__((address_space(3)))

typedef unsigned int u32x4 __attribute__((ext_vector_type(4)));
typedef int          i32x8 __attribute__((ext_vector_type(8)));
typedef int          i32x4 __attribute__((ext_vector_type(4)));

// log(exp(x) + exp(y)) — values stay in ~[-400, 50]; NEG is -100, no infs.
__device__ __forceinline__ float laddexp(float x, float y) {
    float m = fmaxf(x, y);
    float d = fminf(x, y) - m;           // <= 0
    return m + __logf(1.0f + __expf(d));
}

// log(exp(x)+exp(y)+exp(z))
__device__ __forceinline__ float lse3(float x, float y, float z) {
    float m = fmaxf(fmaxf(x, y), z);
    return m + __logf(__expf(x - m) + __expf(y - m) + __expf(z - m));
}

// Delete-chain: fD[0]=NEG; fD[k] = logaddexp(c[k], d[k] + fD[k-1]), k=1..128.
// Lane l owns k in {4l+1..4l+4}. Log-affine-map parallel scan across the wave
// (wave32): in linear domain y = C + D*y_prev composes, so in log domain
// (c2,d2)∘(c1,d1) = (logaddexp(c2, d2+c1), d2+d1).
__device__ __forceinline__ void dchain(int lane, const float c[4], const float dco[4],
                                       float fDn[4]) {
    // compose the lane's 4 steps:  V -> laddexp(C, D + V)
    float C = c[0], D = dco[0];
#pragma unroll
    for (int j = 1; j < 4; ++j) {
        C = laddexp(c[j], dco[j] + C);
        D = D + dco[j];
    }
    // inclusive Hillis-Steele scan of map composition (branch-free: cndmask)
#pragma unroll
    for (int s = 1; s < 32; s <<= 1) {
        float Co = __shfl_up(C, s, 32);
        float Do = __shfl_up(D, s, 32);
        float Cn = laddexp(C, D + Co);   // apply earlier map first
        float Dn = D + Do;
        C = (lane >= s) ? Cn : C;
        D = (lane >= s) ? Dn : D;
    }
    // value entering this lane: fD[4*lane] = (maps of lanes 0..l-1)(fD[0]=NEG)
    float Ce = __shfl_up(C, 1, 32);
    float De = __shfl_up(D, 1, 32);
    float V = (lane == 0) ? NEGV : laddexp(Ce, De + NEGV);
#pragma unroll
    for (int j = 0; j < 4; ++j) {
        V = laddexp(c[j], dco[j] + V);
        fDn[j] = V;
    }
}

// One wave32 per batch element. blockDim = 32, gridDim = B = 512.
__global__ __launch_bounds__(32) void phmm_fwd_kernel(
    const int*   __restrict__ tokens,  // (B, 256)
    const float* __restrict__ trans,   // (B, 129, 7)
    const float* __restrict__ emis,    // (B, 128, 4)
    float*       __restrict__ nll_out) // (B,)
{
    const int L = 256;
    const int b = blockIdx.x;
    const int lane = threadIdx.x;      // 0..31 (wave32)

    __shared__ int   toks[256];
    __shared__ float aLDS[904];        // (K+1)*7 = 903 floats of transition block

    const int*   gtok = tokens + b * L;
    const float* ab   = trans + b * 129 * 7;
    const float* eb   = emis + b * 128 * 4;

#if defined(__AMDGCN__) && __has_builtin(__builtin_amdgcn_tensor_load_to_lds)
    // --- Tensor Data Mover: DMA the 903-float transition block into LDS ---
    {
        unsigned long long ga = (unsigned long long)(const void*)ab;
        unsigned lds_off = (unsigned)(unsigned long long)(AS3 float*)aLDS;
        u32x4 g0;
        g0.x = 1u;                                            // count=1, user D#
        g0.y = lds_off;                                       // lds_addr (bytes)
        g0.z = (unsigned)(ga & 0xFFFFFFFFu);                  // global_addr[31:0]
        g0.w = (unsigned)((ga >> 32) & 0x01FFFFFFu) | (2u << 30); // addr[56:32] | type=2
        i32x8 g1;
        g1[0] = (int)(2u << 16);       // workgroup_mask=0, data_size=2 (4 bytes)
        g1[1] = (int)(903u << 16);     // tensor_dim0[15:0]=903 (abar_addr=0)
        g1[2] = (int)(1u << 16);       // tensor_dim0 hi=0 | tensor_dim1=1
        g1[3] = (int)(903u << 16);     // tensor_dim1 hi=0 | tile_dim0=903
        g1[4] = 1;                     // tile_dim1=1, tile_dim2=0
        g1[5] = 903;                   // tensor_dim0_stride[31:0]
        g1[6] = 0;                     // stride hi | tensor_dim1_stride lo
        g1[7] = 0;
        i32x4 gz4 = {0, 0, 0, 0};
        i32x8 gz8 = {0, 0, 0, 0, 0, 0, 0, 0};
        __builtin_amdgcn_tensor_load_to_lds(g0, g1, gz4, gz4, gz8, 0);
    }
    const float* asrc = (const float*)aLDS;   // coefficients come from LDS
#else
    const float* asrc = ab;                   // fallback: straight from global
#endif

#if defined(__AMDGCN__) && __has_builtin(__builtin_amdgcn_global_load_async_to_lds_b32)
    // --- CDNA5 async global->LDS staging of the token stream ---
#pragma unroll
    for (int u = 0; u < 8; ++u) {
        int i = lane + 32 * u;
        __builtin_amdgcn_global_load_async_to_lds_b32(
            (AS1 int*)(unsigned long long)(gtok + i),
            (AS3 int*)(&toks[i]), 0, 0);
    }
    asm volatile("s_wait_asynccnt 0x0" ::: "memory");
#else
    for (int i = lane; i < L; i += 32) toks[i] = gtok[i];
#endif

#if defined(__AMDGCN__) && __has_builtin(__builtin_amdgcn_tensor_load_to_lds)
    asm volatile("s_wait_tensorcnt 0x0" ::: "memory");
#endif

    // ---- preload loop-invariant coefficients into registers ----
    float aM2M[4], aI2M[4], aD2M[4], aM2D[4], aD2D[4], aM2I[4], aI2I[4];
    float4 em4[4];
#pragma unroll
    for (int j = 0; j < 4; ++j) {
        const int r = 4 * lane + j;            // 0..127 : row k-1 for this lane's k=r+1
        const float* a0 = asrc + r * 7;
        aM2M[j] = a0[0];
        aM2D[j] = a0[2];
        aI2M[j] = a0[3];
        aD2M[j] = a0[5];
        aD2D[j] = a0[6];
        const float* a1 = asrc + (r + 1) * 7;  // row k for fI update
        aM2I[j] = a1[1];
        aI2I[j] = a1[4];
        em4[j]  = *(const float4*)(eb + r * 4); // emission row k-1, all 4 symbols
    }
    const float a0M2I = asrc[1];          // a[0][M2I]
    const float a0I2I = asrc[4];          // a[0][I2I]
    const float aKM2M = asrc[128 * 7 + 0];
    const float aKI2M = asrc[128 * 7 + 3];
    const float aKD2M = asrc[128 * 7 + 5];

    // ---- state: lane l holds k in {4l+1..4l+4}; index 0 kept as scalars ----
    float fM[4], fI[4], fD[4];
    float fM0 = 0.0f, fI0 = NEGV;         // fD[0] is always NEG
#pragma unroll
    for (int j = 0; j < 4; ++j) { fM[j] = NEGV; fI[j] = NEGV; }

    // initial D-chain on the fM init state
    {
        float pM = __shfl_up(fM[3], 1, 32);
        float c[4];
        c[0] = aM2D[0] + ((lane == 0) ? fM0 : pM);
        c[1] = aM2D[1] + fM[0];
        c[2] = aM2D[2] + fM[1];
        c[3] = aM2D[3] + fM[2];
        dchain(lane, c, aD2D, fD);
    }

    // ---- column loop ----
    for (int t = 0; t < L; ++t) {
        const int sym = toks[t];

        // neighbor (k-1) state values
        float pM = __shfl_up(fM[3], 1, 32);
        float pI = __shfl_up(fI[3], 1, 32);
        float pD = __shfl_up(fD[3], 1, 32);
        float prevM[4], prevI[4], prevD[4];
        prevM[0] = (lane == 0) ? fM0 : pM;
        prevI[0] = (lane == 0) ? fI0 : pI;
        prevD[0] = (lane == 0) ? NEGV : pD;   // fD[0] == NEG always
        prevM[1] = fM[0]; prevM[2] = fM[1]; prevM[3] = fM[2];
        prevI[1] = fI[0]; prevI[2] = fI[1]; prevI[3] = fI[2];
        prevD[1] = fD[0]; prevD[2] = fD[1]; prevD[3] = fD[2];

        // fM_new[k] = em[k-1,sym] + lse3(aM2M+fM[k-1], aI2M+fI[k-1], aD2M+fD[k-1])
        float fMn[4];
#pragma unroll
        for (int j = 0; j < 4; ++j) {
            float e = (sym == 0) ? em4[j].x
                    : (sym == 1) ? em4[j].y
                    : (sym == 2) ? em4[j].z : em4[j].w;
            fMn[j] = e + lse3(aM2M[j] + prevM[j],
                              aI2M[j] + prevI[j],
                              aD2M[j] + prevD[j]);
        }

        // fI_new[k] = LOG_QUARTER + logaddexp(a[k,M2I]+fM[k], a[k,I2I]+fI[k])
        float fIn[4];
#pragma unroll
        for (int j = 0; j < 4; ++j)
            fIn[j] = LOG_QUARTER + laddexp(aM2I[j] + fM[j], aI2I[j] + fI[j]);
        const float fI0n = LOG_QUARTER + laddexp(a0M2I + fM0, a0I2I + fI0);

        // fD_new = d-chain over fM_new (fM_new[0] == NEG)
        float pMn = __shfl_up(fMn[3], 1, 32);
        float c[4];
        c[0] = aM2D[0] + ((lane == 0) ? NEGV : pMn);
        c[1] = aM2D[1] + fMn[0];
        c[2] = aM2D[2] + fMn[1];
        c[3] = aM2D[3] + fMn[2];
        dchain(lane, c, aD2D, fD);

        // commit
#pragma unroll
        for (int j = 0; j < 4; ++j) { fM[j] = fMn[j]; fI[j] = fIn[j]; }
        fM0 = NEGV;
        fI0 = fI0n;
    }

    // final: -lse3(fM[K]+a[K,M2M], fI[K]+a[K,I2M], fD[K]+a[K,D2M]); k=128 is lane 31, j=3
    float v = lse3(fM[3] + aKM2M, fI[3] + aKI2M, fD[3] + aKD2M);
    if (lane == 31) nll_out[b] = -v;
}

// KLD + final mean reduction: out = mean_b(nll_b + kld_b)
__global__ __launch_bounds__(256) void finalize_kernel(
    const float* __restrict__ nll,      // (512,)
    const float* __restrict__ mus,      // (512,16)
    const float* __restrict__ logvars,  // (512,16)
    float*       __restrict__ out)      // (1,)
{
    __shared__ float red[256];
    const int tid = threadIdx.x;
    float acc = 0.0f;
    for (int b = tid; b < 512; b += 256) {
        float s = 0.0f;
#pragma unroll
        for (int e = 0; e < 16; ++e) {
            float mu = mus[b * 16 + e];
            float lv = logvars[b * 16 + e];
            s += 1.0f + lv - mu * mu - __expf(lv);
        }
        acc += nll[b] + (-0.5f * s);
    }
    red[tid] = acc;
    __syncthreads();
    for (int s = 128; s > 0; s >>= 1) {
        if (tid < s) red[tid] += red[tid + s];
        __syncthreads();
    }
    if (tid == 0) out[0] = red[0] * (1.0f / 512.0f);
}

extern "C" void kernel_launch(void* const* d_in, const int* in_sizes, int n_in,
                              void* d_out, int out_size, void* d_ws, size_t ws_size,
                              hipStream_t stream) {
    const int*   tokens  = (const int*)d_in[0];    // (512,256) int32
    const float* trans   = (const float*)d_in[1];  // (512,129,7) f32
    const float* emis    = (const float*)d_in[2];  // (512,128,4) f32
    const float* mus     = (const float*)d_in[3];  // (512,16) f32
    const float* logvars = (const float*)d_in[4];  // (512,16) f32
    float* nll = (float*)d_ws;                     // 512 floats of scratch

    phmm_fwd_kernel<<<512, 32, 0, stream>>>(tokens, trans, emis, nll);
    finalize_kernel<<<1, 256, 0, stream>>>(nll, mus, logvars, (float*)d_out);
}